// DetectionLoss_31490700215086
// MI455X (gfx1250) — compile-verified
//
#include <hip/hip_runtime.h>
#include <hip/hip_bf16.h>

#define B_     32
#define P_     16384
#define T_     64
#define CHUNKS 8
#define CP     2048     // preds per chunk
#define TPB    256
#define NW     (TPB/32) // waves per block in kernel 1
#define WPAD   9        // padded stride for wave-partial tables (conflict-free)

// ---- CDNA5 async global->LDS DMA path (guarded; fallback = plain b128 copy)
#if __has_builtin(__builtin_amdgcn_global_load_async_to_lds_b128) && \
    __has_builtin(__builtin_amdgcn_s_wait_asynccnt)
#define USE_ASYNC 1
#else
#define USE_ASYNC 0
#endif

#define AS1 __attribute__((address_space(1)))
#define AS3 __attribute__((address_space(3)))
typedef int v4i __attribute__((vector_size(16)));   // matches builtin param type

__device__ __forceinline__ void copy16_to_lds(const float* g, float* l) {
#if USE_ASYNC
  __builtin_amdgcn_global_load_async_to_lds_b128((AS1 v4i*)g, (AS3 v4i*)l, 0, 0);
#else
  *(float4*)l = *(const float4*)g;
#endif
}
__device__ __forceinline__ void async_join() {
#if USE_ASYNC
  __builtin_amdgcn_s_wait_asynccnt(0);
#endif
}

// ---------------------------------------------------------------------------
// Kernel 1: per (chunk, batch): IoU max/argmax of 2048 preds vs all 64 targets
// ---------------------------------------------------------------------------
__global__ __launch_bounds__(TPB) void k_iou_argmax(
    const float* __restrict__ preds, const float* __restrict__ targets,
    float* __restrict__ part_iou, int* __restrict__ part_idx)
{
  __shared__ __align__(16) float s_pred[CP * 5];   // 40960 B raw pred floats
  __shared__ __align__(16) float s_traw[T_ * 4];   // raw targets
  __shared__ float s_tgt[T_ * 5];                  // x1,y1,x2,y2,area
  __shared__ float s_wiou[T_ * WPAD];              // per-(target,wave) partials
  __shared__ int   s_widx[T_ * WPAD];

  const int tid  = threadIdx.x;
  const int wave = tid >> 5;
  const int lane = tid & 31;
  const int cx   = blockIdx.x;   // chunk
  const int b    = blockIdx.y;   // batch

  // Async-DMA the contiguous raw chunk (2048 preds * 5 floats = 2560 x 16B)
  const float* gsrc = preds + ((size_t)b * P_ + (size_t)cx * CP) * 5;
#pragma unroll
  for (int i = 0; i < 10; ++i) {
    int u = i * TPB + tid;                         // 16B unit index 0..2559
    copy16_to_lds(gsrc + u * 4, s_pred + u * 4);
  }
  if (tid < T_) {                                  // 64 x 16B target records
    copy16_to_lds(targets + ((size_t)b * T_ + tid) * 4, s_traw + tid * 4);
  }
  async_join();
  __syncthreads();

  if (tid < T_) {
    float x = s_traw[tid*4+0], y = s_traw[tid*4+1];
    float w = s_traw[tid*4+2], h = s_traw[tid*4+3];
    s_tgt[tid*5+0] = x; s_tgt[tid*5+1] = y;
    s_tgt[tid*5+2] = x + w; s_tgt[tid*5+3] = y + h; s_tgt[tid*5+4] = w * h;
  }
  __syncthreads();

  // Each thread keeps 8 preds resident in VGPRs, reused across 64 targets.
  float px1[8], py1[8], px2[8], py2[8], pa[8];
#pragma unroll
  for (int i = 0; i < 8; ++i) {
    int L = tid + i * TPB;
    float x1 = s_pred[L*5+0], y1 = s_pred[L*5+1];
    float w  = s_pred[L*5+2], h  = s_pred[L*5+3];
    px1[i] = x1; py1[i] = y1; px2[i] = x1 + w; py2[i] = y1 + h; pa[i] = w * h;
  }

  const int gbase = cx * CP;
  for (int t = 0; t < T_; ++t) {
    float gx1 = s_tgt[t*5+0], gy1 = s_tgt[t*5+1];
    float gx2 = s_tgt[t*5+2], gy2 = s_tgt[t*5+3], ga = s_tgt[t*5+4];
    float bi = -1.0f; int bx = 0;
#pragma unroll
    for (int i = 0; i < 8; ++i) {
      float xa = fmaxf(gx1, px1[i]);
      float ya = fmaxf(gy1, py1[i]);
      float xb = fminf(gx2, px2[i]);
      float yb = fminf(gy2, py2[i]);
      float dx = fmaxf(xb - xa, 0.0f);
      float dy = fmaxf(yb - ya, 0.0f);
      float inter = dx * dy;
      float uni = pa[i] + ga - inter;                  // > 0 always here
      float iou = inter * __builtin_amdgcn_rcpf(uni);  // inter==0 -> 0 exactly
      if (iou > bi) { bi = iou; bx = gbase + tid + i * TPB; }  // first-index on tie
    }
    // wave32 reduce, barrier-free (prefer lower index on equal IoU)
#pragma unroll
    for (int off = 16; off; off >>= 1) {
      float oi = __shfl_down(bi, off, 32);
      int   ox = __shfl_down(bx, off, 32);
      if (oi > bi || (oi == bi && ox < bx)) { bi = oi; bx = ox; }
    }
    if (lane == 0) { s_wiou[t * WPAD + wave] = bi; s_widx[t * WPAD + wave] = bx; }
  }
  __syncthreads();            // single barrier for all 64 targets

  if (tid < T_) {             // parallel final fold: thread t folds 8 wave entries
    float fi = s_wiou[tid * WPAD]; int fx = s_widx[tid * WPAD];
#pragma unroll
    for (int w2 = 1; w2 < NW; ++w2) {
      float oi = s_wiou[tid * WPAD + w2]; int ox = s_widx[tid * WPAD + w2];
      if (oi > fi || (oi == fi && ox < fx)) { fi = oi; fx = ox; }
    }
    int o = (b * T_ + tid) * CHUNKS + cx;
    part_iou[o] = fi; part_idx[o] = fx;
  }
}

// ---------------------------------------------------------------------------
// Kernel 2: per batch: fold chunk partials, build positive-pred bitmask,
// sorted positive index list (== stable argsort of 1-conf_target), bbox loss.
// ---------------------------------------------------------------------------
__global__ __launch_bounds__(512) void k_select(
    const float* __restrict__ preds, const float* __restrict__ targets,
    const float* __restrict__ part_iou, const int* __restrict__ part_idx,
    unsigned* __restrict__ mask_ws, int* __restrict__ nvals,
    float* __restrict__ bbox)
{
  __shared__ unsigned s_mask[512];
  __shared__ unsigned s_scan[512];
  __shared__ int s_sel[T_];
  __shared__ float s_r2[2];
  const int tid = threadIdx.x;
  const int b   = blockIdx.x;

  s_mask[tid] = 0u;
  __syncthreads();

  if (tid < T_) {
    float bi = -1.0f; int bx = 0;
    for (int c = 0; c < CHUNKS; ++c) {   // ascending chunks => first index on tie
      int o = (b * T_ + tid) * CHUNKS + c;
      float oi = part_iou[o]; int ox = part_idx[o];
      if (oi > bi) { bi = oi; bx = ox; }
    }
    if (bi > 0.5f) atomicOr(&s_mask[bx >> 5], 1u << (bx & 31));
  }
  __syncthreads();

  unsigned myw = s_mask[tid];
  unsigned cnt = __popc(myw);
  s_scan[tid] = cnt;
  __syncthreads();
  for (int off = 1; off < 512; off <<= 1) {        // inclusive scan
    unsigned add = (tid >= off) ? s_scan[tid - off] : 0u;
    __syncthreads();
    s_scan[tid] += add;
    __syncthreads();
  }
  unsigned total = s_scan[511];                    // n (<= 64)
  unsigned e = s_scan[tid] - cnt;                  // exclusive prefix
  unsigned w = myw;
  while (w) {                                      // emit set bits in order
    unsigned bit = (unsigned)__builtin_ctz(w);
    if (e < T_) s_sel[e] = (int)(tid * 32u + bit);
    ++e;
    w &= w - 1u;
  }
  __syncthreads();

  float val = 0.0f;
  if (tid < T_) {
    if (tid < (int)total) {                        // valid slot: SSE vs target t
      int p = s_sel[tid];
      const float* pr = preds + ((size_t)b * P_ + p) * 5;   // raw x1,y1,w,h
      const float* tg = targets + ((size_t)b * T_ + tid) * 4;
#pragma unroll
      for (int k = 0; k < 4; ++k) { float d = pr[k] - tg[k]; val += d * d; }
    }
#pragma unroll
    for (int off = 16; off; off >>= 1) val += __shfl_down(val, off, 32);
    if ((tid & 31) == 0) s_r2[tid >> 5] = val;
  }
  __syncthreads();
  if (tid == 0) {
    float s = s_r2[0] + s_r2[1];
    float nf = (float)total;
    bbox[b]  = s / fmaxf(nf * 4.0f, 1.0f);
    nvals[b] = (int)total;
  }
  mask_ws[b * 512 + tid] = myw;
}

// ---------------------------------------------------------------------------
// Kernel 3: BCE-with-logits conf loss, fixed-slot block partials (deterministic)
// ---------------------------------------------------------------------------
__global__ __launch_bounds__(256) void k_confloss(
    const float* __restrict__ preds, const unsigned* __restrict__ mask_ws,
    float* __restrict__ conf_part)
{
  __shared__ float s8[8];
  const int tid = threadIdx.x;
  const int b   = blockIdx.y;
  const int p   = blockIdx.x * 256 + tid;

  float x = preds[((size_t)b * P_ + p) * 5 + 4];
  unsigned m = mask_ws[b * 512 + (p >> 5)];
  int tg = (int)((m >> (p & 31)) & 1u);
  // loss = tg*softplus(-x) + (1-tg)*softplus(x); softplus(z)=max(z,0)+log1p(exp(-|z|))
  float ax = fabsf(x);
  float l = log1pf(__expf(-ax)) + (tg ? fmaxf(-x, 0.0f) : fmaxf(x, 0.0f));
#pragma unroll
  for (int off = 16; off; off >>= 1) l += __shfl_down(l, off, 32);
  if ((tid & 31) == 0) s8[tid >> 5] = l;
  __syncthreads();
  if (tid == 0) {
    float s = 0.0f;
#pragma unroll
    for (int i = 0; i < 8; ++i) s += s8[i];
    conf_part[b * 64 + blockIdx.x] = s;
  }
}

// ---------------------------------------------------------------------------
// Kernel 4: single wave32 finalize
// ---------------------------------------------------------------------------
__global__ __launch_bounds__(32) void k_final(
    const float* __restrict__ conf_part, const int* __restrict__ nvals,
    const float* __restrict__ bbox, float* __restrict__ out)
{
  const int b = threadIdx.x;     // 0..31 == batch
  float s = 0.0f;
  for (int i = 0; i < 64; ++i) s += conf_part[b * 64 + i];
  float conf = s * (1.0f / (float)P_);
  float per  = (nvals[b] > 0) ? (bbox[b] + conf) : 0.0f;
#pragma unroll
  for (int off = 16; off; off >>= 1) per += __shfl_down(per, off, 32);
  if (b == 0) out[0] = per * (1.0f / (float)B_);
}

// ---------------------------------------------------------------------------
extern "C" void kernel_launch(void* const* d_in, const int* in_sizes, int n_in,
                              void* d_out, int out_size, void* d_ws, size_t ws_size,
                              hipStream_t stream)
{
  const float* preds   = (const float*)d_in[0];   // (32,16384,5)
  const float* targets = (const float*)d_in[1];   // (32,64,4)
  char* ws = (char*)d_ws;
  // workspace layout (all regions fully written before being read)
  float*    part_iou  = (float*)(ws);               // 32*64*8 f32 = 65536 B
  int*      part_idx  = (int*)  (ws + 65536);       // 65536 B
  unsigned* mask      = (unsigned*)(ws + 131072);   // 32*512 u32 = 65536 B
  int*      nvals     = (int*)  (ws + 196608);      // 128 B
  float*    bbox      = (float*)(ws + 196736);      // 128 B
  float*    conf_part = (float*)(ws + 196864);      // 32*64 f32 = 8192 B
  float*    out       = (float*)d_out;

  dim3 g1(CHUNKS, B_);
  k_iou_argmax<<<g1, TPB, 0, stream>>>(preds, targets, part_iou, part_idx);
  k_select<<<B_, 512, 0, stream>>>(preds, targets, part_iou, part_idx, mask, nvals, bbox);
  dim3 g3(P_ / 256, B_);
  k_confloss<<<g3, 256, 0, stream>>>(preds, mask, conf_part);
  k_final<<<1, 32, 0, stream>>>(conf_part, nvals, bbox, out);
}